// SpatialAttention_3186865734227
// MI455X (gfx1250) — compile-verified
//
#include <hip/hip_runtime.h>

// Sliding-window attention (w=16, W=33) + fused LayerNorm, fp32,
// on CDNA5 V_WMMA_F32_16X16X4_F32 (native fp32 matrix pipe).
//
// x: (2, 16384, 256) f32 ; gamma,beta: (256,) f32 ; out: (2,16384,256) f32
//
// One wave32 (one workgroup) handles 16 consecutive queries:
//   - stage the 48-row x band (rows q0-16 .. q0+31, zero-padded at sequence
//     edges) into LDS once: 96x global_load_b128 -> ds_store_b128
//   - S(16x48) = Q(16x256) * Kband(48x256)^T   (3 acc tiles x 64 k-chunks,
//     all fragments from LDS; K-tile-1 rows == Q rows so B==A there)
//   - window mask -> -inf ; zero-padded rows give score 0 (zero K vector)
//   - softmax per row (shfl_xor reductions within 16-lane halves)
//   - P staged via LDS (C-layout -> A-layout)
//   - O(16x256) = P(16x48) * Vband(48x256)     (16 d tiles x 12 k-chunks)
//   - fused LayerNorm over D=256 with per-row shuffle-reduced stats.
//
// Zero-filling at staging makes every inner loop check-free (no EXEC ops,
// no cndmasks). Row stride padded to 260 floats so pass-1 B reads (lane
// stride = one row) hit 16 distinct banks; P stride padded to 52.

typedef float v2f __attribute__((ext_vector_type(2)));
typedef float v4f __attribute__((ext_vector_type(4)));
typedef float v8f __attribute__((ext_vector_type(8)));

#define SEQ_N   16384
#define DIM     256
#define RS      260     // padded LDS band row stride (floats), 1040B (16B-mult)
#define PS      52      // padded P-staging row stride (floats)

__global__ __launch_bounds__(32) void swattn_ln_kernel(
    const float* __restrict__ x, const float* __restrict__ gamma,
    const float* __restrict__ beta, float* __restrict__ out) {
  __shared__ float band[48 * RS];   // 49,920 B
  __shared__ float pst[16 * PS];    //  3,328 B

  const int lane = threadIdx.x & 31;
  const int tile = blockIdx.x;                 // 2048 tiles
  const int b    = tile >> 10;                 // SEQ_N/16 = 1024 tiles/batch
  const int q0   = (tile & 1023) << 4;
  const int n16  = lane & 15;                  // B/C/D col index or A row index
  const int hi   = lane >> 4;                  // K split for A/B, M+8 for C/D

  const float* xb = x + (long)b * SEQ_N * DIM;
  float*       ob = out + (long)b * SEQ_N * DIM;

  // ---- stage 48-row band into LDS, zero-fill out-of-range rows ----
  const int g0 = q0 - 16;
#pragma unroll 4
  for (int i = 0; i < 96; ++i) {
    const int row = i >> 1, half = i & 1;
    const int grow = g0 + row;                 // wave-uniform -> scalar branch
    v4f d = v4f{};
    if (grow >= 0 && grow < SEQ_N)
      d = *(const v4f*)(xb + (long)grow * DIM + half * 128 + lane * 4);
    *(v4f*)(&band[row * RS + half * 128 + lane * 4]) = d;
  }
  __syncthreads();   // single-wave workgroup: hardware NOP, orders LDS

  // ---------------- pass 1: S = Q * Kband^T  (16x48) ----------------
  v8f sc[3] = {v8f{}, v8f{}, v8f{}};
  const float* qrow = &band[(16 + n16) * RS];  // A row m = n16  (band row 16+m)
  const float* k0   = &band[n16 * RS];         // B col n -> band row n
  const float* k2   = &band[(32 + n16) * RS];  // B col n -> band row 32+n
  for (int kc = 0; kc < 64; ++kc) {
    const int koff = kc * 4 + hi * 2;          // K = koff, koff+1
    v2f a  = *(const v2f*)(qrow + koff);
    v2f b0 = *(const v2f*)(k0 + koff);
    v2f b2 = *(const v2f*)(k2 + koff);
    sc[0] = __builtin_amdgcn_wmma_f32_16x16x4_f32(false, a, false, b0,
                                                  (short)0, sc[0], false, false);
    sc[1] = __builtin_amdgcn_wmma_f32_16x16x4_f32(false, a, false, a,
                                                  (short)0, sc[1], false, false);
    sc[2] = __builtin_amdgcn_wmma_f32_16x16x4_f32(false, a, false, b2,
                                                  (short)0, sc[2], false, false);
  }

  // -------- mask + softmax (rows spread over 16-lane halves) --------
  const float scale = 0.0625f;  // 256^-0.5
#pragma unroll
  for (int v = 0; v < 8; ++v) {
    const int m = v + 8 * hi;   // query row held by this VGPR in this lane
    float mx = -3.0e38f;
#pragma unroll
    for (int t = 0; t < 3; ++t) {
      const int j = 16 * t + n16;                      // band column
      const bool valid = (j >= m) && (j <= m + 32);    // |i-j|<=16 window
      float val = valid ? sc[t][v] * scale : -3.0e38f;
      sc[t][v] = val;
      mx = fmaxf(mx, val);
    }
#pragma unroll
    for (int off = 1; off < 16; off <<= 1)
      mx = fmaxf(mx, __shfl_xor(mx, off, 32));
    float sum = 0.0f;
#pragma unroll
    for (int t = 0; t < 3; ++t) {
      float e = __expf(sc[t][v] - mx);                 // masked -> 0
      sc[t][v] = e;
      sum += e;
    }
#pragma unroll
    for (int off = 1; off < 16; off <<= 1)
      sum += __shfl_xor(sum, off, 32);
    const float inv = 1.0f / sum;
#pragma unroll
    for (int t = 0; t < 3; ++t) sc[t][v] *= inv;
  }

  // stage P (C-layout) into LDS, reload in A-layout
#pragma unroll
  for (int t = 0; t < 3; ++t)
#pragma unroll
    for (int v = 0; v < 8; ++v)
      pst[(v + 8 * hi) * PS + 16 * t + n16] = sc[t][v];
  __syncthreads();   // hardware NOP, orders LDS for the compiler

  v2f af[12];
  {
    const float* prow = &pst[n16 * PS];                // A row m = n16
#pragma unroll
    for (int kc = 0; kc < 12; ++kc)
      af[kc] = *(const v2f*)(prow + kc * 4 + hi * 2);  // 8B-aligned LDS read
  }

  // ------------ pass 2: O = P * Vband  (16x256) + LN stats ------------
  v8f ot[16];
  v8f rsum = v8f{}, rsq = v8f{};
#pragma unroll
  for (int d = 0; d < 16; ++d) {
    v8f c = v8f{};
#pragma unroll
    for (int kc = 0; kc < 12; ++kc) {
      const int r = kc * 4 + hi * 2;                   // band rows r, r+1
      const float* vp = &band[r * RS + d * 16 + n16];
      v2f bf;
      bf.x = vp[0];
      bf.y = vp[RS];
      c = __builtin_amdgcn_wmma_f32_16x16x4_f32(false, af[kc], false, bf,
                                                (short)0, c, false, false);
    }
    ot[d] = c;
    rsum += c;
    rsq += c * c;
  }

  // per-row mean / rstd (reduce across the 16 lanes holding each row)
  float mean[8], rstd[8];
#pragma unroll
  for (int v = 0; v < 8; ++v) {
    float s = rsum[v], s2 = rsq[v];
#pragma unroll
    for (int off = 1; off < 16; off <<= 1) {
      s += __shfl_xor(s, off, 32);
      s2 += __shfl_xor(s2, off, 32);
    }
    const float mu = s * (1.0f / 256.0f);
    mean[v] = mu;
    rstd[v] = rsqrtf(s2 * (1.0f / 256.0f) - mu * mu + 1e-5f);
  }

  // ---------------- normalize + store ----------------
#pragma unroll
  for (int d = 0; d < 16; ++d) {
    const float g  = gamma[d * 16 + n16];
    const float be = beta[d * 16 + n16];
#pragma unroll
    for (int v = 0; v < 8; ++v) {
      const int m = v + 8 * hi;
      const float val = (ot[d][v] - mean[v]) * rstd[v] * g + be;
      ob[(long)(q0 + m) * DIM + d * 16 + n16] = val;
    }
  }
}

extern "C" void kernel_launch(void* const* d_in, const int* in_sizes, int n_in,
                              void* d_out, int out_size, void* d_ws, size_t ws_size,
                              hipStream_t stream) {
  (void)in_sizes; (void)n_in; (void)d_ws; (void)ws_size; (void)out_size;
  const float* x     = (const float*)d_in[0];
  const float* gamma = (const float*)d_in[1];
  const float* beta  = (const float*)d_in[2];
  float* out = (float*)d_out;
  // 2 batches * 1024 query tiles = 2048 one-wave workgroups
  dim3 grid(2048), block(32);
  swattn_ln_kernel<<<grid, block, 0, stream>>>(x, gamma, beta, out);
}